// GDN_5592047420067
// MI455X (gfx1250) — compile-verified
//
#include <hip/hip_runtime.h>
#include <math.h>

// Problem constants (match reference)
#define Bb  64
#define Nn  512
#define Ff  64
#define Dd  64
#define Kk  20
#define BN  (Bb * Nn)      // 32768
#define NK  (Nn * Kk)      // 10240
#define NEGV (-1.0e9f)

typedef __attribute__((ext_vector_type(2))) float v2f;
typedef __attribute__((ext_vector_type(8))) float v8f;

// ---------------------------------------------------------------------------
// 1) emb row norms
__global__ void norms_k(const float* __restrict__ emb, float* __restrict__ norms) {
  int n = blockIdx.x * blockDim.x + threadIdx.x;
  if (n >= Nn) return;
  const float* e = emb + (size_t)n * Dd;
  float s = 0.f;
#pragma unroll
  for (int d = 0; d < Dd; ++d) s += e[d] * e[d];
  norms[n] = sqrtf(s);
}

// ---------------------------------------------------------------------------
// 2) cosine similarity row + top-K (stable: ties -> lowest index, like lax.top_k)
//    Writes int topk to ws, and learned_graph (as floats) to out_graph.
__global__ void __launch_bounds__(128) cos_topk_k(const float* __restrict__ emb,
                                                  const float* __restrict__ norms,
                                                  int* __restrict__ topk,
                                                  float* __restrict__ out_graph) {
  const int i = blockIdx.x;        // row (node)
  const int t = threadIdx.x;       // 128 threads
  __shared__ float ei[Dd];
  __shared__ float cv[Nn];
  if (t < Dd) ei[t] = emb[(size_t)i * Dd + t];
  __syncthreads();
  const float ni = norms[i];
  for (int j = t; j < Nn; j += 128) {
    const float* ej = emb + (size_t)j * Dd;
    float dot = 0.f;
#pragma unroll
    for (int d = 0; d < Dd; ++d) dot += ei[d] * ej[d];
    cv[j] = dot / (ni * norms[j]);
  }
  __syncthreads();
  // wave 0 performs iterative arg-max selection (20 rounds)
  if (t < 32) {
    for (int k = 0; k < Kk; ++k) {
      float best = -3.0e38f;
      int   bj   = 0x7fffffff;
      for (int j = t; j < Nn; j += 32) {
        float v = cv[j];
        if (v > best) { best = v; bj = j; }   // ascending scan -> lowest index kept
      }
#pragma unroll
      for (int off = 16; off; off >>= 1) {
        float ov = __shfl_xor(best, off, 32);
        int   oj = __shfl_xor(bj, off, 32);
        if (ov > best || (ov == best && oj < bj)) { best = ov; bj = oj; }
      }
      // all lanes agree on bj now
      if (t == 0) {
        topk[i * Kk + k]            = bj;
        out_graph[i * Kk + k]       = (float)bj;  // gated_j
        out_graph[NK + i * Kk + k]  = (float)i;   // gated_i
      }
      cv[bj] = -3.0e38f;   // all lanes store same value: benign, orders per-lane
    }
  }
}

// ---------------------------------------------------------------------------
// 3) xl = x @ lin_w.T via V_WMMA_F32_16X16X4_F32 (fp32 matrix cores).
//    Block = 128 threads = 4 waves; block owns 16 rows of x; wave w owns
//    output D-tile [w*16, w*16+16). K=64 -> 16 WMMA steps of K=4.
__global__ void __launch_bounds__(128) gemm_xl_wmma(const float* __restrict__ x,
                                                    const float* __restrict__ w,
                                                    float* __restrict__ xl) {
  const int wave = threadIdx.x >> 5;
  const int lane = threadIdx.x & 31;
  const int half = lane >> 4;     // 0: lanes 0-15, 1: lanes 16-31
  const int l    = lane & 15;
  const int m0   = blockIdx.x << 4;
  const int d0   = wave << 4;

  // A 16x4 layout: lanes 0-15 hold (K0,K1) of row M=l; lanes 16-31 hold (K2,K3)
  const float* xrow = x + (size_t)(m0 + l) * Ff;
  // B 4x16 layout (mirror): lane picks column N=l; b = B[k][n] = lin_w[n][k]
  const float* wrow = w + (size_t)(d0 + l) * Ff;

  v8f c = {};
#pragma unroll
  for (int kk = 0; kk < 16; ++kk) {
    const int kb = (kk << 2) + (half << 1);
    v2f a = *(const v2f*)(xrow + kb);
    v2f b = *(const v2f*)(wrow + kb);
    c = __builtin_amdgcn_wmma_f32_16x16x4_f32(false, a, false, b,
                                              (short)0, c, false, false);
  }
  // C/D layout: VGPR r -> M = r + 8*half, N = l
  float* o = xl + (size_t)m0 * Dd + d0;
#pragma unroll
  for (int r = 0; r < 8; ++r)
    o[(size_t)(r + (half << 3)) * Dd + l] = c[r];
}

// ---------------------------------------------------------------------------
// 4) per-node attention scores
__global__ void scores_k(const float* __restrict__ xl, const float* __restrict__ emb,
                         const float* __restrict__ att_i, const float* __restrict__ att_j,
                         const float* __restrict__ att_em_i, const float* __restrict__ att_em_j,
                         float* __restrict__ si, float* __restrict__ sj) {
  int n = blockIdx.x * blockDim.x + threadIdx.x;
  if (n >= BN) return;
  const float* xr = xl + (size_t)n * Dd;
  const float* er = emb + (size_t)(n & (Nn - 1)) * Dd;
  float a = 0.f, b = 0.f;
#pragma unroll 8
  for (int d = 0; d < Dd; ++d) {
    float xv = xr[d], ev = er[d];
    a += xv * att_i[d] + ev * att_em_i[d];
    b += xv * att_j[d] + ev * att_em_j[d];
  }
  si[n] = a;
  sj[n] = b;
}

// ---------------------------------------------------------------------------
// 5) norm_emb = softmax(emb, axis=1)
__global__ void softmax_emb_k(const float* __restrict__ emb, float* __restrict__ nemb) {
  int n = blockIdx.x * blockDim.x + threadIdx.x;
  if (n >= Nn) return;
  const float* e = emb + (size_t)n * Dd;
  float m = -3.0e38f;
#pragma unroll
  for (int d = 0; d < Dd; ++d) m = fmaxf(m, e[d]);
  float s = 0.f;
#pragma unroll
  for (int d = 0; d < Dd; ++d) s += expf(e[d] - m);
  float inv = 1.f / s;
  float* o = nemb + (size_t)n * Dd;
#pragma unroll
  for (int d = 0; d < Dd; ++d) o[d] = expf(e[d] - m) * inv;
}

// ---------------------------------------------------------------------------
// 6) attention message passing: one wave32 per destination node.
//    Candidates: e<20 -> topk[i][e] (masked if == i), e==20 -> self loop.
__global__ void __launch_bounds__(256) attn_k(const float* __restrict__ xl,
                                              const float* __restrict__ si,
                                              const float* __restrict__ sj,
                                              const int* __restrict__ topk,
                                              const float* __restrict__ bias,
                                              float* __restrict__ out) {
  const int wid  = (blockIdx.x * blockDim.x + threadIdx.x) >> 5; // dst node (global)
  const int lane = threadIdx.x & 31;
  if (wid >= BN) return;
  const int b = wid >> 9;          // batch
  const int i = wid & (Nn - 1);    // node within graph

  float alpha = -3.0e38f;
  int   srcg  = 0;
  bool  valid = false;
  if (lane <= Kk) {
    const int srcl = (lane < Kk) ? topk[i * Kk + lane] : i;
    srcg  = (b << 9) + srcl;
    valid = (lane == Kk) || (srcl != i);
    float a = si[wid] + sj[srcg];
    a = (a > 0.f) ? a : 0.2f * a;          // leaky_relu(0.2)
    alpha = valid ? a : NEGV;              // where(valid, ., NEG)
  }
  // wave max
  float amax = alpha;
#pragma unroll
  for (int off = 16; off; off >>= 1) amax = fmaxf(amax, __shfl_xor(amax, off, 32));
  // exp & sum
  float ex = (lane <= Kk && valid) ? expf(alpha - amax) : 0.f;
  float denom = ex;
#pragma unroll
  for (int off = 16; off; off >>= 1) denom += __shfl_xor(denom, off, 32);
  const float attw = ex / denom;

  // weighted gather-sum: each lane owns dims {2*lane, 2*lane+1}
  const float2* xl2 = (const float2*)xl;
  float accx = 0.f, accy = 0.f;
#pragma unroll
  for (int e = 0; e <= Kk; ++e) {
    const float we = __shfl(attw, e, 32);
    const int   se = __shfl(srcg, e, 32);
    const float2 v = xl2[(size_t)se * (Dd / 2) + lane];
    accx += we * v.x;
    accy += we * v.y;
  }
  const float2 bo = ((const float2*)bias)[lane];
  float2 r; r.x = accx + bo.x; r.y = accy + bo.y;
  ((float2*)out)[(size_t)wid * (Dd / 2) + lane] = r;
}

// ---------------------------------------------------------------------------
// 7) per-channel (d = idx & 63) partial sums for BatchNorm stats (coalesced).
//    Strides (256, 32768) are multiples of 64, so each thread's channel is fixed.
__global__ void __launch_bounds__(256) bn_partial_k(const float* __restrict__ z,
                                                    float* __restrict__ part) {
  const int t = threadIdx.x;
  float s = 0.f, s2 = 0.f;
  for (int idx = blockIdx.x * 256 + t; idx < BN * Dd; idx += 128 * 256) {
    float v = z[idx];
    s += v; s2 += v * v;
  }
  __shared__ float S[256], S2[256];
  S[t] = s; S2[t] = s2;
  __syncthreads();
  if (t < 64) {
    s  = S[t]  + S[t + 64]  + S[t + 128]  + S[t + 192];
    s2 = S2[t] + S2[t + 64] + S2[t + 128] + S2[t + 192];
    part[blockIdx.x * 64 + t]        = s;
    part[8192 + blockIdx.x * 64 + t] = s2;
  }
}

__global__ void bn_final_k(const float* __restrict__ part,
                           float* __restrict__ mean, float* __restrict__ inv) {
  const int d = threadIdx.x;
  if (d >= Dd) return;
  float s = 0.f, s2 = 0.f;
  for (int k = 0; k < 128; ++k) { s += part[k * 64 + d]; s2 += part[8192 + k * 64 + d]; }
  const float m = s / (float)BN;
  const float v = s2 / (float)BN - m * m;
  mean[d] = m;
  inv[d]  = rsqrtf(v + 1e-5f);
}

// ---------------------------------------------------------------------------
// 8) apply bn1 + ReLU + embedding gate (in place)
__global__ void apply1_k(float* __restrict__ z, const float* __restrict__ mean,
                         const float* __restrict__ inv, const float* __restrict__ g,
                         const float* __restrict__ bta, const float* __restrict__ nemb) {
  const int idx = blockIdx.x * blockDim.x + threadIdx.x;
  if (idx >= BN * Dd) return;
  const int d = idx & (Dd - 1);
  const int n = (idx >> 6) & (Nn - 1);
  float v = (z[idx] - mean[d]) * inv[d] * g[d] + bta[d];
  v = fmaxf(v, 0.f);
  z[idx] = v * nemb[n * Dd + d];
}

// ---------------------------------------------------------------------------
// 9) apply bn2 + ReLU + output projection
__global__ void final_k(const float* __restrict__ h, const float* __restrict__ mean,
                        const float* __restrict__ inv, const float* __restrict__ g,
                        const float* __restrict__ bta, const float* __restrict__ ow,
                        const float* __restrict__ ob, float* __restrict__ y) {
  const int r = blockIdx.x * blockDim.x + threadIdx.x;
  if (r >= BN) return;
  const float* hr = h + (size_t)r * Dd;
  float acc = 0.f;
#pragma unroll 8
  for (int d = 0; d < Dd; ++d) {
    float v = (hr[d] - mean[d]) * inv[d] * g[d] + bta[d];
    v = fmaxf(v, 0.f);
    acc += v * ow[d];
  }
  y[r] = acc + ob[0];
}

// ---------------------------------------------------------------------------
extern "C" void kernel_launch(void* const* d_in, const int* in_sizes, int n_in,
                              void* d_out, int out_size, void* d_ws, size_t ws_size,
                              hipStream_t stream) {
  const float* data     = (const float*)d_in[0];   // [B,N,F] == [BN,64]
  const float* emb      = (const float*)d_in[1];   // [N,D]
  const float* lin_w    = (const float*)d_in[2];   // [D,F]
  const float* att_i    = (const float*)d_in[3];
  const float* att_j    = (const float*)d_in[4];
  const float* att_em_i = (const float*)d_in[5];
  const float* att_em_j = (const float*)d_in[6];
  const float* gnn_bias = (const float*)d_in[7];
  const float* bn1_g    = (const float*)d_in[8];
  const float* bn1_b    = (const float*)d_in[9];
  const float* bn2_g    = (const float*)d_in[10];
  const float* bn2_b    = (const float*)d_in[11];
  const float* out_w    = (const float*)d_in[12];
  const float* out_b    = (const float*)d_in[13];
  float* out = (float*)d_out;                      // [BN] y, then [2*NK] graph

  // workspace layout (floats)
  float* ws      = (float*)d_ws;
  float* xl      = ws;                       // 2,097,152
  float* outbuf  = ws + 2097152;             // 2,097,152 (gnn out -> h in place)
  float* si      = ws + 4194304;             // 32768
  float* sj      = si + BN;                  // 32768
  float* norms   = sj + BN;                  // 512
  float* nemb    = norms + Nn;               // 32768
  float* bn_mean = nemb + Nn * Dd;           // 64 (reused bn1 then bn2)
  float* bn_inv  = bn_mean + Dd;             // 64
  float* part    = bn_inv + Dd;              // 16384
  int*   topk    = (int*)(part + 16384);     // 10240 ints

  // 1) norms
  norms_k<<<2, 256, 0, stream>>>(emb, norms);
  // 2) cosine + top-k + learned_graph output
  cos_topk_k<<<Nn, 128, 0, stream>>>(emb, norms, topk, out + BN);
  // 3) xl = x @ lin_w.T  (WMMA fp32)
  gemm_xl_wmma<<<BN / 16, 128, 0, stream>>>(data, lin_w, xl);
  // 4) attention scores
  scores_k<<<BN / 256, 256, 0, stream>>>(xl, emb, att_i, att_j, att_em_i, att_em_j, si, sj);
  // 5) softmax(emb, axis=1)
  softmax_emb_k<<<2, 256, 0, stream>>>(emb, nemb);
  // 6) message passing (wave per dst node)
  attn_k<<<(BN * 32) / 256, 256, 0, stream>>>(xl, si, sj, topk, gnn_bias, outbuf);
  // 7) bn1 stats
  bn_partial_k<<<128, 256, 0, stream>>>(outbuf, part);
  bn_final_k<<<1, 64, 0, stream>>>(part, bn_mean, bn_inv);
  // 8) bn1 + relu + embedding gate (in place)
  apply1_k<<<(BN * Dd) / 256, 256, 0, stream>>>(outbuf, bn_mean, bn_inv, bn1_g, bn1_b, nemb);
  // 9) bn2 stats
  bn_partial_k<<<128, 256, 0, stream>>>(outbuf, part);
  bn_final_k<<<1, 64, 0, stream>>>(part, bn_mean, bn_inv);
  // 10) bn2 + relu + out projection -> y
  final_k<<<BN / 256, 256, 0, stream>>>(outbuf, bn_mean, bn_inv, bn2_g, bn2_b, out_w, out_b, out);

  (void)in_sizes; (void)n_in; (void)out_size; (void)ws_size;
}